// MyModel_85083302134416
// MI455X (gfx1250) — compile-verified
//
#include <hip/hip_runtime.h>
#include <math.h>

// Problem constants (from reference)
#define N_SENT  100000
#define D_DIM   2070
#define K_PAD   2072          // D padded to multiple of 4 (WMMA K-step)
#define N_BAGS  20000
#define N_REL   53
#define NCOL    64            // padded output cols: 0..52 = classifier, 53 = att logit, 54..63 = 0
#define M_TILES (N_SENT / 16) // 6250, exact

typedef __attribute__((ext_vector_type(2))) float v2f;
typedef __attribute__((ext_vector_type(8))) float v8f;

// ---------------------------------------------------------------------------
// K0: build augmented transposed weights Wt[64][K_PAD]
//     Wt[n][k] = cls_w[k][n] (n<53) | att_w[k] (n==53) | 0 otherwise / k>=2070
// ---------------------------------------------------------------------------
__global__ void pack_w(const float* __restrict__ att_w,
                       const float* __restrict__ cls_w,
                       float* __restrict__ Wt) {
  int idx = blockIdx.x * 256 + threadIdx.x;
  if (idx >= NCOL * K_PAD) return;
  int n = idx / K_PAD;
  int k = idx % K_PAD;
  float v = 0.0f;
  if (k < D_DIM) {
    if (n < N_REL)       v = cls_w[(size_t)k * N_REL + n];
    else if (n == N_REL) v = att_w[k];
  }
  Wt[idx] = v;
}

// ---------------------------------------------------------------------------
// K1: Y[100000][64] = X[100000][2070] @ Wt^T  via V_WMMA_F32_16X16X4_F32
//     one wave per 16-row M-tile; 4 N-tiles (64 cols) accumulated per wave
// ---------------------------------------------------------------------------
__global__ __launch_bounds__(128) void gemm_xw(const float* __restrict__ X,
                                               const float* __restrict__ Wt,
                                               float* __restrict__ Y) {
  const int lane = threadIdx.x & 31;
  const int wave = threadIdx.x >> 5;
  const int tile = blockIdx.x * 4 + wave;   // wave-uniform guard: EXEC stays all-1s
  if (tile >= M_TILES) return;

  const int m  = lane & 15;   // A: row-in-tile / B: col-in-tile / D: N index
  const int hi = lane >> 4;   // K half-select per ISA fragment layout
  const int o0 = 2 * hi;      // lane's K sub-offset within a K=4 step

  const float* __restrict__ arow = X  + (size_t)(tile * 16 + m) * D_DIM;
  const float* __restrict__ b0   = Wt + (size_t)(0 * 16 + m) * K_PAD;
  const float* __restrict__ b1   = Wt + (size_t)(1 * 16 + m) * K_PAD;
  const float* __restrict__ b2   = Wt + (size_t)(2 * 16 + m) * K_PAD;
  const float* __restrict__ b3   = Wt + (size_t)(3 * 16 + m) * K_PAD;

  v8f c0 = {}, c1 = {}, c2 = {}, c3 = {};

  // main K loop: 517 steps of K=4 covering k = 0..2067 (all loads 8B-aligned)
  for (int kk = 0; kk < 2068; kk += 4) {
    const int o = kk + o0;
    float2 af = *reinterpret_cast<const float2*>(arow + o);
    v2f a; a.x = af.x; a.y = af.y;

    float2 f0 = *reinterpret_cast<const float2*>(b0 + o);
    float2 f1 = *reinterpret_cast<const float2*>(b1 + o);
    float2 f2 = *reinterpret_cast<const float2*>(b2 + o);
    float2 f3 = *reinterpret_cast<const float2*>(b3 + o);
    v2f vb0; vb0.x = f0.x; vb0.y = f0.y;
    v2f vb1; vb1.x = f1.x; vb1.y = f1.y;
    v2f vb2; vb2.x = f2.x; vb2.y = f2.y;
    v2f vb3; vb3.x = f3.x; vb3.y = f3.y;

    c0 = __builtin_amdgcn_wmma_f32_16x16x4_f32(false, a, false, vb0, (short)0, c0, false, false);
    c1 = __builtin_amdgcn_wmma_f32_16x16x4_f32(false, a, false, vb1, (short)0, c1, false, false);
    c2 = __builtin_amdgcn_wmma_f32_16x16x4_f32(false, a, false, vb2, (short)0, c2, false, false);
    c3 = __builtin_amdgcn_wmma_f32_16x16x4_f32(false, a, false, vb3, (short)0, c3, false, false);
  }

  // tail step: k = 2068..2071 (A guarded: k>=2070 -> 0; Wt is zero-padded)
  {
    const int o = 2068 + o0;
    v2f a;
    a.x = (o     < D_DIM) ? arow[o]     : 0.0f;
    a.y = (o + 1 < D_DIM) ? arow[o + 1] : 0.0f;

    float2 f0 = *reinterpret_cast<const float2*>(b0 + o);
    float2 f1 = *reinterpret_cast<const float2*>(b1 + o);
    float2 f2 = *reinterpret_cast<const float2*>(b2 + o);
    float2 f3 = *reinterpret_cast<const float2*>(b3 + o);
    v2f vb0; vb0.x = f0.x; vb0.y = f0.y;
    v2f vb1; vb1.x = f1.x; vb1.y = f1.y;
    v2f vb2; vb2.x = f2.x; vb2.y = f2.y;
    v2f vb3; vb3.x = f3.x; vb3.y = f3.y;

    c0 = __builtin_amdgcn_wmma_f32_16x16x4_f32(false, a, false, vb0, (short)0, c0, false, false);
    c1 = __builtin_amdgcn_wmma_f32_16x16x4_f32(false, a, false, vb1, (short)0, c1, false, false);
    c2 = __builtin_amdgcn_wmma_f32_16x16x4_f32(false, a, false, vb2, (short)0, c2, false, false);
    c3 = __builtin_amdgcn_wmma_f32_16x16x4_f32(false, a, false, vb3, (short)0, c3, false, false);
  }

  // store D: VGPR v -> M = v + 8*hi, N = nt*16 + m
  float* yb = Y + (size_t)(tile * 16) * NCOL;
#pragma unroll
  for (int v = 0; v < 8; ++v) {
    const int mr = v + 8 * hi;
    yb[mr * NCOL + 0 * 16 + m] = c0[v];
    yb[mr * NCOL + 1 * 16 + m] = c1[v];
    yb[mr * NCOL + 2 * 16 + m] = c2[v];
    yb[mr * NCOL + 3 * 16 + m] = c3[v];
  }
}

// ---------------------------------------------------------------------------
// K2: per-bag softmax over logits Y[:,53]; writes p_i back into Y[:,53].
//     Handles arbitrarily large bags via grid-stride; deterministic LDS tree.
// ---------------------------------------------------------------------------
__global__ __launch_bounds__(256) void bag_softmax(const long long* __restrict__ scope,
                                                   float* __restrict__ Y) {
  const int b = blockIdx.x;
  const int tid = threadIdx.x;
  const int start = (int)scope[2 * b];
  const int end   = (int)scope[2 * b + 1];

  __shared__ float red[256];

  // 1) max
  float lm = -INFINITY;
  for (int i = start + tid; i < end; i += 256)
    lm = fmaxf(lm, Y[(size_t)i * NCOL + N_REL]);
  red[tid] = lm;
  __syncthreads();
  for (int s = 128; s > 0; s >>= 1) {
    if (tid < s) red[tid] = fmaxf(red[tid], red[tid + s]);
    __syncthreads();
  }
  const float mx = red[0];
  __syncthreads();

  // 2) sum of exp
  float ls = 0.0f;
  for (int i = start + tid; i < end; i += 256)
    ls += expf(Y[(size_t)i * NCOL + N_REL] - mx);
  red[tid] = ls;
  __syncthreads();
  for (int s = 128; s > 0; s >>= 1) {
    if (tid < s) red[tid] += red[tid + s];
    __syncthreads();
  }
  const float z = red[0];
  __syncthreads();

  // 3) write normalized weights in place
  for (int i = start + tid; i < end; i += 256)
    Y[(size_t)i * NCOL + N_REL] = expf(Y[(size_t)i * NCOL + N_REL] - mx) / z;
}

// ---------------------------------------------------------------------------
// K3: out[b][c] = cls_b[c] + sum_{i in bag} p_i * Y[i][c]   (c < 53)
//     1024 threads: 16 row-groups x 64 cols; fixed reduction order.
// ---------------------------------------------------------------------------
__global__ __launch_bounds__(1024) void bag_reduce(const long long* __restrict__ scope,
                                                   const float* __restrict__ Y,
                                                   const float* __restrict__ cls_b,
                                                   float* __restrict__ out) {
  const int b = blockIdx.x;
  const int tid = threadIdx.x;
  const int c = tid & 63;     // column
  const int g = tid >> 6;     // row group 0..15
  const int start = (int)scope[2 * b];
  const int end   = (int)scope[2 * b + 1];

  float acc = 0.0f;
  for (int i = start + g; i < end; i += 16) {
    const float p = Y[(size_t)i * NCOL + N_REL];
    acc += p * Y[(size_t)i * NCOL + c];
  }

  __shared__ float red[1024];
  red[tid] = acc;
  __syncthreads();

  if (g == 0 && c < N_REL) {
    float s = 0.0f;
#pragma unroll
    for (int gg = 0; gg < 16; ++gg) s += red[gg * 64 + c];
    out[(size_t)b * N_REL + c] = s + cls_b[c];
  }
}

// ---------------------------------------------------------------------------
extern "C" void kernel_launch(void* const* d_in, const int* in_sizes, int n_in,
                              void* d_out, int out_size, void* d_ws, size_t ws_size,
                              hipStream_t stream) {
  const float*     X     = (const float*)d_in[0];      // [100000, 2070]
  const float*     att_w = (const float*)d_in[1];      // [2070]
  const float*     cls_w = (const float*)d_in[2];      // [2070, 53]
  const float*     cls_b = (const float*)d_in[3];      // [53]
  const long long* scope = (const long long*)d_in[4];  // [20000, 2] int64
  float*           out   = (float*)d_out;              // [20000, 53]

  // workspace layout: Wt[64*K_PAD] then Y[N_SENT*64]  (~26.2 MB total)
  float* Wt = (float*)d_ws;
  float* Y  = Wt + (size_t)NCOL * K_PAD;

  pack_w     <<<(NCOL * K_PAD + 255) / 256, 256, 0, stream>>>(att_w, cls_w, Wt);
  gemm_xw    <<<(M_TILES + 3) / 4,          128, 0, stream>>>(X, Wt, Y);
  bag_softmax<<<N_BAGS,                     256, 0, stream>>>(scope, Y);
  bag_reduce <<<N_BAGS,                    1024, 0, stream>>>(scope, Y, cls_b, out);
}